// TNeck_12695923327656
// MI455X (gfx1250) — compile-verified
//
#include <hip/hip_runtime.h>
#include <hip/hip_bf16.h>

typedef float v2f __attribute__((ext_vector_type(2)));
typedef float v8f __attribute__((ext_vector_type(8)));

static __host__ int cdiv_host(long a, long b) { return (int)((a + b - 1) / b); }

// ---------------- utility ----------------
__global__ void k_zero(float* __restrict__ p, int n) {
    int i = blockIdx.x * blockDim.x + threadIdx.x;
    if (i < n) p[i] = 0.f;
}

// deg = segment_sum(edge_weight, src)
__global__ void k_deg_scatter(const int* __restrict__ src, const float* __restrict__ ew,
                              float* __restrict__ deg, int E) {
    int e = blockIdx.x * blockDim.x + threadIdx.x;
    if (e < E) atomicAdd(&deg[src[e]], ew[e]);
}

__global__ void k_dinv(const float* __restrict__ deg, float* __restrict__ dinv, int n) {
    int i = blockIdx.x * blockDim.x + threadIdx.x;
    if (i < n) {
        float d = deg[i];
        dinv[i] = (d > 0.f) ? rsqrtf(d) : 0.f;
    }
}

__global__ void k_wnorm(const int* __restrict__ src, const int* __restrict__ dst,
                        const float* __restrict__ ew, const float* __restrict__ dinv,
                        float* __restrict__ wn, int E) {
    int e = blockIdx.x * blockDim.x + threadIdx.x;
    if (e < E) wn[e] = dinv[src[e]] * ew[e] * dinv[dst[e]];
}

// Ax[:,0:96]=[z|x] (T0), Ax[:,96:192]=0 (T1), Ax[:,192:288]=-T0 (T2 seed)
__global__ void k_init_basis_x(const float* __restrict__ z, const float* __restrict__ x,
                               float* __restrict__ Ax, int N) {
    int t = blockIdx.x * blockDim.x + threadIdx.x;
    if (t >= N * 96) return;
    int i = t / 96, c = t - i * 96;
    float v = (c < 64) ? z[(long)i * 64 + c] : x[(long)i * 32 + (c - 64)];
    float* row = Ax + (long)i * 288;
    row[c] = v;
    row[96 + c] = 0.f;
    row[192 + c] = -v;
}

// Ah[:,0:64]=h (T0), Ah[:,64:128]=0, Ah[:,128:192]=-h
__global__ void k_init_basis_h(const float* __restrict__ h, float* __restrict__ Ah, int N) {
    int t = blockIdx.x * blockDim.x + threadIdx.x;
    if (t >= N * 64) return;
    int i = t >> 6, c = t & 63;
    float v = h[(long)i * 64 + c];
    float* row = Ah + (long)i * 192;
    row[c] = v;
    row[64 + c] = 0.f;
    row[128 + c] = -v;
}

// tout[dst] += factor * wn[e] * tin[src]   (float4 per thread; hits L2)
__global__ void k_prop_scatter(const int* __restrict__ src, const int* __restrict__ dst,
                               const float* __restrict__ wn,
                               const float* __restrict__ tin, float* __restrict__ tout,
                               int ld, int groups /*C/4*/, float factor, int E) {
    int t = blockIdx.x * blockDim.x + threadIdx.x;
    if (t >= E * groups) return;
    int e = t / groups;
    int g = (t - e * groups) << 2;
    float w = factor * wn[e];
    const float4 v = *reinterpret_cast<const float4*>(tin + (long)src[e] * ld + g);
    float* d = tout + (long)dst[e] * ld + g;
    atomicAdd(d + 0, w * v.x);
    atomicAdd(d + 1, w * v.y);
    atomicAdd(d + 2, w * v.z);
    atomicAdd(d + 3, w * v.w);
}

// ---------------- WMMA f32 GEMM: out[N,64] (+)= A[N,Kdim] @ W[Kdim,64] ----------------
// 8 waves / block; block stages W[:, col0:col0+16] in LDS once; each wave owns one
// 16x16 output tile and runs V_WMMA_F32_16X16X4_F32 over K.
#define GEMM_WAVES 8
__global__ void __launch_bounds__(256)
k_gemm_wmma(const float* __restrict__ A, int lda, int Kdim,
            const float* __restrict__ W,
            const float* __restrict__ b0, const float* __restrict__ b1,
            float* __restrict__ out, int accumulate, int N) {
    extern __shared__ float ldsW[];  // [Kdim][16] row-major
    const int tid = threadIdx.x;
    const int lane = tid & 31;
    const int wave = tid >> 5;
    const int n = lane & 15;       // A row within tile / C column
    const int khalf = lane >> 4;   // selects K pair {0,1} or {2,3}
    const int col0 = blockIdx.x * 16;
    const int rowoff = khalf ? 8 : 0;  // C fragment: M = v + 8*(lane>=16)

    // cooperative stage of the 16-column weight slice
    for (int idx = tid; idx < Kdim * 16; idx += 256) {
        int k = idx >> 4, c = idx & 15;
        ldsW[idx] = W[(long)k * 64 + col0 + c];
    }
    __syncthreads();

    const long row0 = ((long)blockIdx.y * GEMM_WAVES + wave) * 16;

    v8f c;
    if (accumulate) {
#pragma unroll
        for (int v = 0; v < 8; ++v) {
            long r = row0 + v + rowoff;
            c[v] = (r < N) ? out[r * 64 + col0 + n] : 0.f;
        }
    } else {
#pragma unroll
        for (int v = 0; v < 8; ++v) c[v] = 0.f;
    }

    long arow = row0 + n;
    if (arow >= N) arow = N - 1;  // harmless clamp; stores are guarded
    const float* Ar = A + arow * lda;

#pragma unroll 4
    for (int kk = 0; kk < Kdim; kk += 4) {
        int ka = kk + khalf * 2;
        v2f a, b;
        a.x = Ar[ka];
        a.y = Ar[ka + 1];
        b.x = ldsW[ka * 16 + n];        // 64 consecutive words over the wave: no bank conflicts
        b.y = ldsW[(ka + 1) * 16 + n];
        c = __builtin_amdgcn_wmma_f32_16x16x4_f32(
                /*neg_a=*/false, a, /*neg_b=*/false, b,
                /*c_mod=*/(short)0, c, /*reuse_a=*/false, /*reuse_b=*/false);
    }

    if (!accumulate && b0) {
        float bias = b0[col0 + n] + (b1 ? b1[col0 + n] : 0.f);
#pragma unroll
        for (int v = 0; v < 8; ++v) c[v] += bias;
    }
#pragma unroll
    for (int v = 0; v < 8; ++v) {
        long r = row0 + v + rowoff;
        if (r < N) out[r * 64 + col0 + n] = c[v];
    }
}

// Z=sigmoid(Z), R=sigmoid(R); seed Ahr basis with T0 = prev_h * R
__global__ void k_gate_zr(float* __restrict__ Z, float* __restrict__ R,
                          const float* __restrict__ prev_h, float* __restrict__ Ahr, int N) {
    int t = blockIdx.x * blockDim.x + threadIdx.x;
    if (t >= N * 64) return;
    int i = t >> 6, c = t & 63;
    float zv = 1.f / (1.f + expf(-Z[t]));
    float rv = 1.f / (1.f + expf(-R[t]));
    Z[t] = zv;
    R[t] = rv;
    float v = prev_h[t] * rv;
    float* row = Ahr + (long)i * 192;
    row[c] = v;
    row[64 + c] = 0.f;
    row[128 + c] = -v;
}

// h = Z*prev_h + (1-Z)*tanh(G)
__global__ void k_final_h(const float* __restrict__ Z, const float* __restrict__ prev_h,
                          const float* __restrict__ G, float* __restrict__ hout, int N) {
    int t = blockIdx.x * blockDim.x + threadIdx.x;
    if (t >= N * 64) return;
    float zv = Z[t];
    hout[t] = zv * prev_h[t] + (1.f - zv) * tanhf(G[t]);
}

// fused[batch[i], 0:64] += z[i]
__global__ void k_graph_emb(const float* __restrict__ z, const int* __restrict__ batch,
                            float* __restrict__ fused, int N) {
    int t = blockIdx.x * blockDim.x + threadIdx.x;
    if (t >= N * 64) return;
    int i = t >> 6, c = t & 63;
    atomicAdd(&fused[(long)batch[i] * 128 + c], z[t]);
}

// fused[b, 64:128] = relu(u @ Wg + bg)
__global__ void k_global_emb(const float* __restrict__ u, const float* __restrict__ Wg,
                             const float* __restrict__ bg, float* __restrict__ fused, int B) {
    int t = blockIdx.x * blockDim.x + threadIdx.x;
    if (t >= B * 64) return;
    int b = t >> 6, co = t & 63;
    float acc = bg[co];
    const float* ur = u + (long)b * 64;
#pragma unroll 8
    for (int k = 0; k < 64; ++k) acc = fmaf(ur[k], Wg[(long)k * 64 + co], acc);
    fused[(long)b * 128 + 64 + co] = fmaxf(acc, 0.f);
}

// ---------------- host ----------------
extern "C" void kernel_launch(void* const* d_in, const int* in_sizes, int n_in,
                              void* d_out, int out_size, void* d_ws, size_t ws_size,
                              hipStream_t stream) {
    const float* x      = (const float*)d_in[0];
    const float* u      = (const float*)d_in[1];
    const float* z      = (const float*)d_in[2];
    const int*   ei     = (const int*)d_in[3];
    const float* ew     = (const float*)d_in[4];
    const int*   batch  = (const int*)d_in[5];
    const float* prev_h = (const float*)d_in[7];
    const float* Wxz = (const float*)d_in[8],  *bxz = (const float*)d_in[9];
    const float* Whz = (const float*)d_in[10], *bhz = (const float*)d_in[11];
    const float* Wxr = (const float*)d_in[12], *bxr = (const float*)d_in[13];
    const float* Whr = (const float*)d_in[14], *bhr = (const float*)d_in[15];
    const float* Wxh = (const float*)d_in[16], *bxh = (const float*)d_in[17];
    const float* Whh = (const float*)d_in[18], *bhh = (const float*)d_in[19];
    const float* Wg  = (const float*)d_in[20], *bg  = (const float*)d_in[21];

    const int N = in_sizes[0] / 32;
    const int E = in_sizes[4];
    const int B = in_sizes[1] / 64;
    const int* src = ei;
    const int* dst = ei + E;

    float* ws = (float*)d_ws;
    long off = 0;
    float* Ax  = ws + off; off += (long)N * 288;
    float* Ah  = ws + off; off += (long)N * 192;
    float* Ahr = ws + off; off += (long)N * 192;
    float* Zb  = ws + off; off += (long)N * 64;
    float* Rb  = ws + off; off += (long)N * 64;
    float* Gh  = ws + off; off += (long)N * 64;
    float* deg = ws + off; off += N;
    float* din = ws + off; off += N;
    float* wn  = ws + off; off += E;

    const int TB = 256;

    // symmetric normalization
    k_zero<<<cdiv_host(N, TB), TB, 0, stream>>>(deg, N);
    k_deg_scatter<<<cdiv_host(E, TB), TB, 0, stream>>>(src, ew, deg, E);
    k_dinv<<<cdiv_host(N, TB), TB, 0, stream>>>(deg, din, N);
    k_wnorm<<<cdiv_host(E, TB), TB, 0, stream>>>(src, dst, ew, din, wn, E);

    // Chebyshev bases for xin and prev_h
    k_init_basis_x<<<cdiv_host((long)N * 96, TB), TB, 0, stream>>>(z, x, Ax, N);
    k_init_basis_h<<<cdiv_host((long)N * 64, TB), TB, 0, stream>>>(prev_h, Ah, N);
    k_prop_scatter<<<cdiv_host((long)E * 24, TB), TB, 0, stream>>>(src, dst, wn, Ax,       Ax + 96,  288, 24, -1.f, E);
    k_prop_scatter<<<cdiv_host((long)E * 24, TB), TB, 0, stream>>>(src, dst, wn, Ax + 96,  Ax + 192, 288, 24, -2.f, E);
    k_prop_scatter<<<cdiv_host((long)E * 16, TB), TB, 0, stream>>>(src, dst, wn, Ah,       Ah + 64,  192, 16, -1.f, E);
    k_prop_scatter<<<cdiv_host((long)E * 16, TB), TB, 0, stream>>>(src, dst, wn, Ah + 64,  Ah + 128, 192, 16, -2.f, E);

    // gate pre-activations via WMMA (weights already [K*cin,64] row-major)
    dim3 gg(4, (unsigned)cdiv_host(N, 16 * GEMM_WAVES));
    size_t lds96 = 288 * 16 * sizeof(float);
    size_t lds64 = 192 * 16 * sizeof(float);
    k_gemm_wmma<<<gg, 256, lds96, stream>>>(Ax, 288, 288, Wxz, bxz, bhz, Zb, 0, N);
    k_gemm_wmma<<<gg, 256, lds64, stream>>>(Ah, 192, 192, Whz, nullptr, nullptr, Zb, 1, N);
    k_gemm_wmma<<<gg, 256, lds96, stream>>>(Ax, 288, 288, Wxr, bxr, bhr, Rb, 0, N);
    k_gemm_wmma<<<gg, 256, lds64, stream>>>(Ah, 192, 192, Whr, nullptr, nullptr, Rb, 1, N);
    k_gemm_wmma<<<gg, 256, lds96, stream>>>(Ax, 288, 288, Wxh, bxh, bhh, Gh, 0, N);

    // gates + candidate basis (data-dependent on R)
    k_gate_zr<<<cdiv_host((long)N * 64, TB), TB, 0, stream>>>(Zb, Rb, prev_h, Ahr, N);
    k_prop_scatter<<<cdiv_host((long)E * 16, TB), TB, 0, stream>>>(src, dst, wn, Ahr,      Ahr + 64,  192, 16, -1.f, E);
    k_prop_scatter<<<cdiv_host((long)E * 16, TB), TB, 0, stream>>>(src, dst, wn, Ahr + 64, Ahr + 128, 192, 16, -2.f, E);
    k_gemm_wmma<<<gg, 256, lds64, stream>>>(Ahr, 192, 192, Whh, nullptr, nullptr, Gh, 1, N);

    // outputs
    float* fused = (float*)d_out;
    float* hout  = fused + (long)B * 128;
    k_final_h<<<cdiv_host((long)N * 64, TB), TB, 0, stream>>>(Zb, prev_h, Gh, hout, N);
    k_zero<<<cdiv_host((long)B * 128, TB), TB, 0, stream>>>(fused, B * 128);
    k_graph_emb<<<cdiv_host((long)N * 64, TB), TB, 0, stream>>>(z, batch, fused, N);
    k_global_emb<<<cdiv_host((long)B * 64, TB), TB, 0, stream>>>(u, Wg, bg, fused, B);
}